// PointNetSetAbstractionSSG_30056181138029
// MI455X (gfx1250) — compile-verified
//
#include <hip/hip_runtime.h>

#define B_   16
#define N_   4096
#define S_   1024
#define K_   32
#define R2_  0.04f
#define EPS_ 1e-5f

typedef float v2f __attribute__((ext_vector_type(2)));
typedef float v8f __attribute__((ext_vector_type(8)));

static __device__ __forceinline__ v8f wmma_f32(v2f a, v2f b, v8f c) {
  // V_WMMA_F32_16X16X4_F32 : D(16x16,f32) = A(16x4,f32) * B(4x16,f32) + C
  return __builtin_amdgcn_wmma_f32_16x16x4_f32(false, a, false, b, (short)0, c,
                                               false, false);
}

// ---------------------------------------------------------------------------
// Kernel 1: farthest point sampling. One workgroup per batch; xyz in LDS.
// Writes new_xyz (first output) directly.
// ---------------------------------------------------------------------------
__global__ __launch_bounds__(256) void fps_kernel(const float* __restrict__ xyz,
                                                  float* __restrict__ newxyz) {
  __shared__ __align__(16) float sx[N_ * 3];
  __shared__ float wval[8];
  __shared__ int   widx[8];
  const int b    = blockIdx.x;
  const int tid  = threadIdx.x;
  const int lane = tid & 31;
  const int wv   = tid >> 5;

  const float* bx = xyz + (size_t)b * N_ * 3;
  for (int i = tid; i < N_ * 3; i += 256) sx[i] = bx[i];
  __syncthreads();

  float ld[16];
#pragma unroll
  for (int j = 0; j < 16; ++j) ld[j] = 1e10f;
  const int base = tid * 16;  // contiguous ownership -> global tie-break by index
  int far = 0;

  for (int i = 0; i < S_; ++i) {
    if (tid == 0) {
      newxyz[((size_t)b * S_ + i) * 3 + 0] = sx[far * 3 + 0];
      newxyz[((size_t)b * S_ + i) * 3 + 1] = sx[far * 3 + 1];
      newxyz[((size_t)b * S_ + i) * 3 + 2] = sx[far * 3 + 2];
    }
    const float cx = sx[far * 3 + 0], cy = sx[far * 3 + 1], cz = sx[far * 3 + 2];
    float best = -1.0f;
    int   bidx = base;
#pragma unroll
    for (int j = 0; j < 16; ++j) {
      const int   p  = base + j;
      const float dx = sx[p * 3 + 0] - cx;
      const float dy = sx[p * 3 + 1] - cy;
      const float dz = sx[p * 3 + 2] - cz;
      const float d  = dx * dx + dy * dy + dz * dz;
      ld[j] = fminf(ld[j], d);
      if (ld[j] > best) { best = ld[j]; bidx = p; }  // strict > keeps first occurrence
    }
    // wave32 reduction (larger dist wins; tie -> smaller index)
#pragma unroll
    for (int off = 16; off > 0; off >>= 1) {
      const float ov = __shfl_xor(best, off, 32);
      const int   oi = __shfl_xor(bidx, off, 32);
      if (ov > best || (ov == best && oi < bidx)) { best = ov; bidx = oi; }
    }
    if (lane == 0) { wval[wv] = best; widx[wv] = bidx; }
    __syncthreads();
    float bv = wval[0];
    int   bi = widx[0];
#pragma unroll
    for (int w = 1; w < 8; ++w) {
      const float v  = wval[w];
      const int   ii = widx[w];
      if (v > bv || (v == bv && ii < bi)) { bv = v; bi = ii; }
    }
    far = bi;
    __syncthreads();
  }
}

// ---------------------------------------------------------------------------
// Kernel 2: ball query. One wave per center; ascending scan picks the first
// K_ in-radius indices (== reference's top_k(-cand) semantics).
// ---------------------------------------------------------------------------
__global__ __launch_bounds__(128) void ball_kernel(const float* __restrict__ xyz,
                                                   const float* __restrict__ newxyz,
                                                   int* __restrict__ idxws) {
  const int lane = threadIdx.x & 31;
  const int g    = blockIdx.x * 4 + (threadIdx.x >> 5);
  const int b    = g >> 10;
  const float cx = newxyz[(size_t)g * 3 + 0];
  const float cy = newxyz[(size_t)g * 3 + 1];
  const float cz = newxyz[(size_t)g * 3 + 2];
  const float* bx = xyz + (size_t)b * N_ * 3;
  int* og = idxws + (size_t)g * K_;

  int cnt = 0, first = -1;
  for (int basep = 0; basep < N_ && cnt < K_; basep += 32) {
    const int   p  = basep + lane;
    const float dx = bx[p * 3 + 0] - cx;
    const float dy = bx[p * 3 + 1] - cy;
    const float dz = bx[p * 3 + 2] - cz;
    const float d2 = dx * dx + dy * dy + dz * dz;
    const bool  in = (d2 <= R2_);
    const unsigned m = (unsigned)__ballot(in);
    if (first < 0 && m) first = basep + __builtin_ctz(m);
    if (in) {
      const int pos = cnt + __popc(m & ((1u << lane) - 1u));
      if (pos < K_) og[pos] = p;
    }
    cnt += __popc(m);
  }
  if (first < 0) first = 0;
  if (lane >= cnt) og[lane] = first;  // fill tail with first index
}

// ---------------------------------------------------------------------------
// Kernel 3: fused group-gather + 3-layer pointwise MLP (WMMA f32) + BN/ReLU
// + max-pool over nsample. One wave per group (32x6 input tile).
// Weights staged in LDS pair-packed ([k/2][col][2]) so every A and B fragment
// is a single ds_load_b64; layer-1 input tile staged per-wave in LDS so all
// fragment builds are branch-free (no exec-mask load divergence).
// ---------------------------------------------------------------------------
__global__ __launch_bounds__(64) void mlp_kernel(
    const float* __restrict__ xyz, const float* __restrict__ pts,
    const float* __restrict__ newxyz, const int* __restrict__ idxws,
    const float* __restrict__ W1, const float* __restrict__ g1,
    const float* __restrict__ bt1, const float* __restrict__ mm1,
    const float* __restrict__ vv1,
    const float* __restrict__ W2, const float* __restrict__ g2,
    const float* __restrict__ bt2, const float* __restrict__ mm2,
    const float* __restrict__ vv2,
    const float* __restrict__ W3, const float* __restrict__ g3,
    const float* __restrict__ bt3, const float* __restrict__ mm3,
    const float* __restrict__ vv3,
    float* __restrict__ feat) {
  // Pair-packed weights: sW[kp * cols + col] holds {W[2kp][col], W[2kp+1][col]}
  __shared__ __align__(16) float sW1[4 * 64 * 2];    //  2 KB (cin padded 6->8)
  __shared__ __align__(16) float sW2[32 * 64 * 2];   // 16 KB
  __shared__ __align__(16) float sW3[32 * 128 * 2];  // 32 KB
  __shared__ __align__(16) float inb[2][32 * 8];     //  2 KB grouped input tiles
  __shared__ __align__(16) float act1[2][32 * 64];   // 16 KB
  __shared__ __align__(16) float act2[2][32 * 64];   // 16 KB
  const int tid  = threadIdx.x;
  const int wv   = tid >> 5;
  const int lane = tid & 31;
  const int half = lane >> 4;  // K / M half-split per ISA f32 WMMA layouts
  const int lr   = lane & 15;

  // ---- stage weights (uniform control flow; blockDim 64 keeps kp uniform) --
  for (int i = tid; i < 4 * 64; i += 64) {
    const int kp = i >> 6, col = i & 63, k = 2 * kp;
    sW1[i * 2 + 0] = (k < 6) ? W1[k * 64 + col] : 0.0f;
    sW1[i * 2 + 1] = (k + 1 < 6) ? W1[(k + 1) * 64 + col] : 0.0f;
  }
  for (int i = tid; i < 32 * 64; i += 64) {
    const int kp = i >> 6, col = i & 63;
    sW2[i * 2 + 0] = W2[(2 * kp) * 64 + col];
    sW2[i * 2 + 1] = W2[(2 * kp + 1) * 64 + col];
  }
  for (int i = tid; i < 32 * 128; i += 64) {
    const int kp = i >> 7, col = i & 127;
    sW3[i * 2 + 0] = W3[(2 * kp) * 128 + col];
    sW3[i * 2 + 1] = W3[(2 * kp + 1) * 128 + col];
  }
  __syncthreads();

  const v2f* vW1 = (const v2f*)sW1;
  const v2f* vW2 = (const v2f*)sW2;
  const v2f* vW3 = (const v2f*)sW3;
  float* ib  = inb[wv];
  float* b0  = act1[wv];
  float* b1b = act2[wv];
  const v2f* vib = (const v2f*)ib;
  const v2f* vb0 = (const v2f*)b0;
  const v2f* vb1 = (const v2f*)b1b;
  const int stride = gridDim.x * 2;

  for (int g = blockIdx.x * 2 + wv; g < B_ * S_; g += stride) {
    const int   b  = g >> 10;
    const float cx = newxyz[(size_t)g * 3 + 0];
    const float cy = newxyz[(size_t)g * 3 + 1];
    const float cz = newxyz[(size_t)g * 3 + 2];
    const int*  gi = idxws + (size_t)g * K_;

    // ---- build 32x8 grouped input tile (one row per lane, branch-free) ----
    {
      const int j = gi[lane];
      const float* pxyz = xyz + ((size_t)b * N_ + j) * 3;
      const float* ppts = pts + ((size_t)b * N_ + j) * 3;
      float* r = ib + lane * 8;
      r[0] = pxyz[0] - cx;
      r[1] = pxyz[1] - cy;
      r[2] = pxyz[2] - cz;
      r[3] = ppts[0];
      r[4] = ppts[1];
      r[5] = ppts[2];
      r[6] = 0.0f;
      r[7] = 0.0f;
    }

    // ---- Layer 1: (32x6) @ W1(6x64), BN+ReLU -> act1 -----------------------
    for (int nt = 0; nt < 4; ++nt) {
      const int   col = 16 * nt + lr;
      const float sc  = g1[col] * rsqrtf(vv1[col] + EPS_);
      const float sh  = bt1[col] - mm1[col] * sc;
      for (int m = 0; m < 2; ++m) {
        const int row = 16 * m + lr;
        v8f acc = {0, 0, 0, 0, 0, 0, 0, 0};
#pragma unroll
        for (int kk = 0; kk < 2; ++kk) {
          const int kp = 2 * kk + half;  // K pair index
          acc = wmma_f32(vib[row * 4 + kp], vW1[kp * 64 + col], acc);
        }
#pragma unroll
        for (int r = 0; r < 8; ++r) {
          const float val = acc[r] * sc + sh;  // D row = 16m + 8*half + r
          b0[(16 * m + 8 * half + r) * 64 + col] = fmaxf(val, 0.0f);
        }
      }
    }

    // ---- Layer 2: (32x64) @ W2(64x64), BN+ReLU -> act2 ---------------------
    for (int nt = 0; nt < 4; ++nt) {
      const int   col = 16 * nt + lr;
      const float sc  = g2[col] * rsqrtf(vv2[col] + EPS_);
      const float sh  = bt2[col] - mm2[col] * sc;
      for (int m = 0; m < 2; ++m) {
        const int row = 16 * m + lr;
        v8f acc = {0, 0, 0, 0, 0, 0, 0, 0};
#pragma unroll 4
        for (int kk = 0; kk < 16; ++kk) {
          const int kp = 2 * kk + half;
          acc = wmma_f32(vb0[row * 32 + kp], vW2[kp * 64 + col], acc);
        }
#pragma unroll
        for (int r = 0; r < 8; ++r) {
          const float val = acc[r] * sc + sh;
          b1b[(16 * m + 8 * half + r) * 64 + col] = fmaxf(val, 0.0f);
        }
      }
    }

    // ---- Layer 3: (32x64) @ W3(64x128), BN+ReLU, max over nsample ----------
    for (int nt = 0; nt < 8; ++nt) {
      const int   col = 16 * nt + lr;
      const float sc  = g3[col] * rsqrtf(vv3[col] + EPS_);
      const float sh  = bt3[col] - mm3[col] * sc;
      float cm = 0.0f;  // post-ReLU values are >= 0
      for (int m = 0; m < 2; ++m) {
        const int row = 16 * m + lr;
        v8f acc = {0, 0, 0, 0, 0, 0, 0, 0};
#pragma unroll 4
        for (int kk = 0; kk < 16; ++kk) {
          const int kp = 2 * kk + half;
          acc = wmma_f32(vb1[row * 32 + kp], vW3[kp * 128 + col], acc);
        }
#pragma unroll
        for (int r = 0; r < 8; ++r)
          cm = fmaxf(cm, fmaxf(acc[r] * sc + sh, 0.0f));  // max over rows in tile
      }
      cm = fmaxf(cm, __shfl_xor(cm, 16, 32));  // combine lane halves (rows r / r+8)
      if (half == 0) feat[(size_t)g * 128 + col] = cm;
    }
  }
}

// ---------------------------------------------------------------------------
extern "C" void kernel_launch(void* const* d_in, const int* in_sizes, int n_in,
                              void* d_out, int out_size, void* d_ws, size_t ws_size,
                              hipStream_t stream) {
  (void)in_sizes; (void)n_in; (void)out_size; (void)ws_size;
  const float* xyz = (const float*)d_in[0];
  const float* pts = (const float*)d_in[1];
  const float* W1  = (const float*)d_in[2];
  const float* g1  = (const float*)d_in[3];
  const float* bt1 = (const float*)d_in[4];
  const float* mm1 = (const float*)d_in[5];
  const float* vv1 = (const float*)d_in[6];
  const float* W2  = (const float*)d_in[7];
  const float* g2  = (const float*)d_in[8];
  const float* bt2 = (const float*)d_in[9];
  const float* mm2 = (const float*)d_in[10];
  const float* vv2 = (const float*)d_in[11];
  const float* W3  = (const float*)d_in[12];
  const float* g3  = (const float*)d_in[13];
  const float* bt3 = (const float*)d_in[14];
  const float* mm3 = (const float*)d_in[15];
  const float* vv3 = (const float*)d_in[16];

  float* out    = (float*)d_out;
  float* newxyz = out;                         // (16,1024,3)
  float* feat   = out + (size_t)B_ * S_ * 3;   // (16,1024,128)
  int*   idxws  = (int*)d_ws;                  // (16,1024,32) ball-query indices

  fps_kernel<<<B_, 256, 0, stream>>>(xyz, newxyz);
  ball_kernel<<<(B_ * S_) / 4, 128, 0, stream>>>(xyz, newxyz, idxws);
  mlp_kernel<<<2048, 64, 0, stream>>>(xyz, pts, newxyz, idxws,
                                      W1, g1, bt1, mm1, vv1,
                                      W2, g2, bt2, mm2, vv2,
                                      W3, g3, bt3, mm3, vv3, feat);
}